// SelfOrganizingMap_20658792694128
// MI455X (gfx1250) — compile-verified
//
#include <hip/hip_runtime.h>

// ---------------- problem constants (from reference) ----------------
#define GSIDE         128
#define G_UNITS       (GSIDE * GSIDE)        // 16384 units
#define DIMS          512
#define NSAMP         4096
#define LRATE         0.1f
#define INV_DENOM     0.5f                   // 1/(2*sigma^2), sigma=1

// ---------------- kernel geometry ----------------
#define NWG           128                    // persistent workgroups (1 per WGP)
#define ROWS_PER_WG   (G_UNITS / NWG)        // 128 rows, 256KB of LDS
#define TPB           256                    // 8 wave32
#define NWAVE         (TPB / 32)
#define ROWS_PER_WAVE (ROWS_PER_WG / NWAVE)  // 16 rows per wave -> WMMA A tile
#define WSTRIDE       516                    // padded LDS row stride (floats): bank-conflict free
#define LW_BYTES      (ROWS_PER_WG * WSTRIDE * 4)   // 264192 B dynamic LDS

typedef float v2f __attribute__((ext_vector_type(2)));
typedef float v8f __attribute__((ext_vector_type(8)));
typedef int   v2i __attribute__((vector_size(8)));   // b64 payload type
typedef __attribute__((address_space(1))) v2i as1_v2i;
typedef __attribute__((address_space(3))) v2i as3_v2i;

// Async global -> LDS DMA (ASYNCcnt path), 8 bytes per lane.
__device__ __forceinline__ void async_copy_b64(const float* gsrc, float* ldst) {
  __builtin_amdgcn_global_load_async_to_lds_b64((as1_v2i*)gsrc, (as3_v2i*)ldst,
                                                0, 0);
}

// Device-wide generation barrier (one arrival thread per WG).
__device__ __forceinline__ void global_barrier(unsigned* cnt, unsigned* gen,
                                               unsigned nwg) {
  __syncthreads();
  if (threadIdx.x == 0) {
    unsigned g = __hip_atomic_load(gen, __ATOMIC_RELAXED,
                                   __HIP_MEMORY_SCOPE_AGENT);
    unsigned arrived = __hip_atomic_fetch_add(cnt, 1u, __ATOMIC_ACQ_REL,
                                              __HIP_MEMORY_SCOPE_AGENT);
    if (arrived == nwg - 1u) {
      __hip_atomic_store(cnt, 0u, __ATOMIC_RELAXED, __HIP_MEMORY_SCOPE_AGENT);
      __hip_atomic_fetch_add(gen, 1u, __ATOMIC_RELEASE,
                             __HIP_MEMORY_SCOPE_AGENT);
    } else {
      while (__hip_atomic_load(gen, __ATOMIC_ACQUIRE,
                               __HIP_MEMORY_SCOPE_AGENT) == g)
        __builtin_amdgcn_s_sleep(1);
    }
  }
  __syncthreads();
}

// Reset per-step BMU slots + barrier state before the persistent kernel runs.
__global__ void som_init(unsigned long long* slots, unsigned* bar) {
  for (int i = 0; i < NSAMP; ++i) slots[i] = ~0ull;
  bar[0] = 0u;   // arrival count
  bar[1] = 0u;   // generation
}

__global__ void __launch_bounds__(TPB, 1)
som_kernel(const float* __restrict__ xin, const float* __restrict__ w0,
           float* __restrict__ wout, unsigned long long* __restrict__ slots,
           unsigned* __restrict__ bar) {
  extern __shared__ float lw[];              // [ROWS_PER_WG][WSTRIDE] weights
  __shared__ float xs[2][DIMS];              // double-buffered sample vector
  __shared__ float s_ww[TPB];                // per-lane partial |w|^2
  __shared__ float s_dot[NWAVE * 16];        // WMMA column-0 extraction
  __shared__ float s_a[ROWS_PER_WG];         // LR * neighborhood per row
  __shared__ unsigned s_bmu;

  const int tid  = threadIdx.x;
  const int wg   = blockIdx.x;
  const int lane = tid & 31;
  const int wv   = tid >> 5;
  const int gRowBase = wg * ROWS_PER_WG;

  // ---- Stage this WG's 128 weight rows into LDS via async DMA (padded rows).
  for (int r = 0; r < ROWS_PER_WG; ++r)
    async_copy_b64(&w0[(size_t)(gRowBase + r) * DIMS + tid * 2],
                   &lw[r * WSTRIDE + tid * 2]);
  async_copy_b64(&xin[tid * 2], &xs[0][tid * 2]);   // prefetch x[0]
  __builtin_amdgcn_s_wait_asynccnt(0);
  __syncthreads();

  // WMMA A-operand addressing per ISA layout (16x4 f32):
  //   lanes 0-15 carry row M=lane, K in {0,1}; lanes 16-31 carry K in {2,3}.
  const int r16  = lane & 15;
  const int cSel = (lane >> 4) << 1;
  const float* wrow = &lw[(wv * ROWS_PER_WAVE + r16) * WSTRIDE];

  for (int n = 0; n < NSAMP; ++n) {
    const int buf = n & 1;
    const float* x = xs[buf];

    // Prefetch next sample into the other buffer (wraps harmlessly at the end).
    async_copy_b64(&xin[(size_t)((n + 1) & (NSAMP - 1)) * DIMS + tid * 2],
                   &xs[buf ^ 1][tid * 2]);
    __builtin_amdgcn_s_wait_asynccnt(1);  // in-order: x[n] (oldest) retired
    __syncthreads();                      // all waves' x slices visible

    // ---- Phase A: BMU search.  argmin of |w|^2 - 2*(w.x)  (|x|^2 constant).
    // w.x for the wave's 16 rows via V_WMMA_F32_16X16X4_F32, x in B column 0;
    // |w|^2 accumulated on the VALU concurrently (WMMA co-executes as TRANS).
    v8f acc = {};
    float ww = 0.0f;
#pragma unroll 8
    for (int k = 0; k < DIMS; k += 4) {
      v2f a = *(const v2f*)&wrow[k + cSel];
      v2f b = *(const v2f*)&x[k + cSel];
      acc = __builtin_amdgcn_wmma_f32_16x16x4_f32(false, a, false, b,
                                                  (short)0, acc, false, false);
      ww = __builtin_fmaf(a.x, a.x, ww);
      ww = __builtin_fmaf(a.y, a.y, ww);
    }
    s_ww[tid] = ww;
    // D column 0: lane 0 holds M=0..7 in v0..v7, lane 16 holds M=8..15.
    if (lane == 0) {
#pragma unroll
      for (int j = 0; j < 8; ++j) s_dot[wv * 16 + j] = acc[j];
    } else if (lane == 16) {
#pragma unroll
      for (int j = 0; j < 8; ++j) s_dot[wv * 16 + 8 + j] = acc[j];
    }
    __syncthreads();

    unsigned long long key = ~0ull;
    if (lane < 16) {
      const unsigned gunit = (unsigned)(gRowBase + wv * ROWS_PER_WAVE + lane);
      const float score =
          (s_ww[tid] + s_ww[tid + 16]) - 2.0f * s_dot[wv * 16 + lane];
      unsigned u = __float_as_uint(score);
      u = (u & 0x80000000u) ? ~u : (u | 0x80000000u);  // order-preserving map
      key = ((unsigned long long)u << 32) | gunit;     // min => first argmin
    }
#pragma unroll
    for (int off = 16; off > 0; off >>= 1) {
      unsigned lo = __shfl_xor((unsigned)key, off, 32);
      unsigned hi = __shfl_xor((unsigned)(key >> 32), off, 32);
      unsigned long long o = ((unsigned long long)hi << 32) | lo;
      key = (o < key) ? o : key;
    }
    if (lane == 0) atomicMin(&slots[n], key);

    // One device-wide barrier per step: all atomicMins land, then everyone
    // reads the winner.  Per-step slots => no reset races, no second barrier.
    global_barrier(&bar[0], &bar[1], NWG);

    // ---- Phase B: neighborhood update of this WG's LDS-resident rows.
    if (tid == 0)
      s_bmu = (unsigned)__hip_atomic_load(&slots[n], __ATOMIC_ACQUIRE,
                                          __HIP_MEMORY_SCOPE_AGENT);
    __syncthreads();
    const int bi = (int)(s_bmu >> 7), bj = (int)(s_bmu & 127u);
    if (tid < ROWS_PER_WG) {
      const int g = gRowBase + tid;
      const float di = (float)((g >> 7) - bi);
      const float dj = (float)((g & 127) - bj);
      s_a[tid] =
          LRATE * __expf(-__builtin_sqrtf(di * di + dj * dj) * INV_DENOM);
    }
    __syncthreads();
    const v2f xv = *(const v2f*)&x[tid * 2];
    for (int r = 0; r < ROWS_PER_WG; ++r) {
      const float a = s_a[r];
      v2f* p = (v2f*)&lw[r * WSTRIDE] + tid;
      v2f w2 = *p;
      w2.x = __builtin_fmaf(a, xv.x - w2.x, w2.x);   // w += a*(x - w)
      w2.y = __builtin_fmaf(a, xv.y - w2.y, w2.y);
      *p = w2;
    }
    __syncthreads();  // updates complete before next step's WMMA reads
  }

  // ---- Final writeout of this WG's rows to d_out.
  for (int r = 0; r < ROWS_PER_WG; ++r)
    *(v2f*)&wout[(size_t)(gRowBase + r) * DIMS + tid * 2] =
        *(const v2f*)&lw[r * WSTRIDE + tid * 2];
}

extern "C" void kernel_launch(void* const* d_in, const int* in_sizes, int n_in,
                              void* d_out, int out_size, void* d_ws,
                              size_t ws_size, hipStream_t stream) {
  const float* xin = (const float*)d_in[0];   // [N, D] input_data
  const float* w0  = (const float*)d_in[1];   // [G, D] weights
  // d_in[2] (grid_coords) is rederived arithmetically from the unit index.
  float* wout = (float*)d_out;

  unsigned long long* slots = (unsigned long long*)d_ws;   // NSAMP u64 keys
  unsigned* bar = (unsigned*)(slots + NSAMP);              // {count, gen}

  (void)in_sizes; (void)n_in; (void)out_size; (void)ws_size;

  (void)hipFuncSetAttribute((const void*)som_kernel,
                            hipFuncAttributeMaxDynamicSharedMemorySize,
                            LW_BYTES);

  som_init<<<1, 1, 0, stream>>>(slots, bar);
  som_kernel<<<NWG, TPB, LW_BYTES, stream>>>(xin, w0, wout, slots, bar);
}